// BlockLinearLayer_41523743818184
// MI455X (gfx1250) — compile-verified
//
#include <hip/hip_runtime.h>

typedef float v2f __attribute__((ext_vector_type(2)));
typedef float v4f __attribute__((ext_vector_type(4)));
typedef float v8f __attribute__((ext_vector_type(8)));

#define ROWS   16                // batch rows per workgroup
#define IN_SIZE 4096
#define OUT_SIZE 1024
#define BLK    32                // block_size (K)
#define NBLK   128               // number of diagonal blocks
#define OPB    8                 // outputs per block (N, padded to 16 in WMMA)
#define NPAD   16                // padded N in Wp
#define CBLKS  16                // blocks staged per chunk
#define CCOLS  (CBLKS * BLK)     // 512 x-columns per chunk
#define XSTR   (CCOLS + 4)       // padded LDS row stride (bank-conflict-free)
#define SSTR   (CBLKS * OPB + 4) // 132, padded out-slab stride
#define WP_ELEMS (NBLK * NPAD * BLK)   // 65536 floats = 256 KB

// ---- pre-pass: expand W[128][8][32] -> Wp[128][16][32], rows j>=8 zeroed ----
__global__ __launch_bounds__(256)
void pad_weights(const float* __restrict__ W, float* __restrict__ Wp)
{
    const int i = blockIdx.x * 256 + threadIdx.x;   // over 65536 elements
    const int k = i & (BLK - 1);
    const int j = (i >> 5) & (NPAD - 1);
    const int n = i >> 9;
    Wp[i] = (j < OPB) ? W[(n * OPB + j) * BLK + k] : 0.0f;
}

// ---- main kernel ----
// PADDED=true : Wsrc = Wp[128][16][32], unconditional B loads
// PADDED=false: Wsrc = W[128][8][32], predicated B loads (fallback, no ws)
template <bool PADDED>
__global__ __launch_bounds__(256)
void block_linear_wmma_f32(const float* __restrict__ x,
                           const float* __restrict__ Wsrc,
                           const float* __restrict__ bias,
                           float* __restrict__ out)
{
    __shared__ float lx[ROWS * XSTR];   // x slab: 16 x 516 floats (~33 KB)
    __shared__ float ls[ROWS * SSTR];   // out slab: 16 x 132 floats (~8.25 KB)

    const int tid  = threadIdx.x;
    const int lane = tid & 31;
    const int wave = tid >> 5;          // 0..7
    const int lrow = lane & 15;         // M index for A; N index for B/C/D
    const int half = lane >> 4;         // lane half selects K pair / M+8
    const size_t row0 = (size_t)blockIdx.x * ROWS;

    const int wstride = PADDED ? (NPAD * BLK) : (OPB * BLK);
    const bool jvalid = (lrow < OPB);

    for (int chunk = 0; chunk < NBLK / CBLKS; ++chunk) {
        // ---- cooperative, fully-coalesced x tile load (streamed: NT) ----
        {
            const int tq = tid & 127;   // float4 index within a row (128/row)
            const int tr = tid >> 7;    // 0..1
#pragma unroll
            for (int p = 0; p < 8; ++p) {
                const int r = p * 2 + tr;
                const v4f v = __builtin_nontemporal_load(
                    reinterpret_cast<const v4f*>(
                        x + (row0 + r) * IN_SIZE + chunk * CCOLS + tq * 4));
                *reinterpret_cast<v4f*>(&lx[r * XSTR + tq * 4]) = v;
            }
        }
        __syncthreads();

        // ---- per-wave: two 16x8 blocks via chained V_WMMA_F32_16X16X4_F32 ----
        v8f acc0, acc1;
#pragma unroll
        for (int i = 0; i < 8; ++i) { acc0[i] = 0.0f; acc1[i] = 0.0f; }

        const int nl0 = wave * 2;            // block-in-chunk
        const int nl1 = nl0 + 1;
        const int n0  = chunk * CBLKS + nl0; // global block id
        const int n1  = n0 + 1;

        // A fragments from LDS: half 0 -> K{4c,4c+1}, half 1 -> K{4c+2,4c+3}
        const float* a0 = &lx[lrow * XSTR + nl0 * BLK + half * 2];
        const float* a1 = &lx[lrow * XSTR + nl1 * BLK + half * 2];
        // B fragments: B[k][j] = W[n][j][k]; j = lrow
        const float* w0 = Wsrc + (size_t)n0 * wstride + lrow * BLK + half * 2;
        const float* w1 = Wsrc + (size_t)n1 * wstride + lrow * BLK + half * 2;

#pragma unroll
        for (int c = 0; c < 8; ++c) {        // K = 32 in chunks of 4
            const v2f av0 = *(const v2f*)(a0 + c * 4);
            const v2f av1 = *(const v2f*)(a1 + c * 4);
            v2f bv0, bv1;
            if (PADDED) {
                bv0 = *(const v2f*)(w0 + c * 4);
                bv1 = *(const v2f*)(w1 + c * 4);
            } else {
                bv0 = (v2f){0.0f, 0.0f};
                bv1 = (v2f){0.0f, 0.0f};
                if (jvalid) {
                    bv0 = *(const v2f*)(w0 + c * 4);
                    bv1 = *(const v2f*)(w1 + c * 4);
                }
            }
            acc0 = __builtin_amdgcn_wmma_f32_16x16x4_f32(
                       false, av0, false, bv0, (short)0, acc0, false, false);
            acc1 = __builtin_amdgcn_wmma_f32_16x16x4_f32(
                       false, av1, false, bv1, (short)0, acc1, false, false);
        }

        // ---- D -> out slab in LDS (only N<8 valid); conflict-free stores ----
        if (jvalid) {
#pragma unroll
            for (int v = 0; v < 8; ++v) {
                const int m = v + half * 8;          // C/D row per ISA layout
                ls[m * SSTR + nl0 * OPB + lrow] = acc0[v];
                ls[m * SSTR + nl1 * OPB + lrow] = acc1[v];
            }
        }
        __syncthreads();

        // ---- coalesced out store with fused bias (streamed: NT store) ----
        {
            const int c4 = lane;                     // 32 float4 = 128 cols
#pragma unroll
            for (int p = 0; p < 2; ++p) {
                const int r = p * 8 + wave;
                v4f v = *reinterpret_cast<const v4f*>(&ls[r * SSTR + c4 * 4]);
                const v4f bb = *reinterpret_cast<const v4f*>(
                    bias + chunk * (CBLKS * OPB) + c4 * 4);
                v = v + bb;
                __builtin_nontemporal_store(v,
                    reinterpret_cast<v4f*>(
                        out + (row0 + r) * OUT_SIZE + chunk * (CBLKS * OPB) + c4 * 4));
            }
        }
        __syncthreads();   // protect ls (and lx) before next chunk overwrites
    }
}

extern "C" void kernel_launch(void* const* d_in, const int* in_sizes, int n_in,
                              void* d_out, int out_size, void* d_ws, size_t ws_size,
                              hipStream_t stream) {
    const float* x  = (const float*)d_in[0];   // [B, 4096]
    const float* W  = (const float*)d_in[1];   // [128, 8, 32]
    const float* b  = (const float*)d_in[2];   // [1024]
    float* out      = (float*)d_out;           // [B, 1024]

    const int batch = in_sizes[0] / IN_SIZE;   // 65536
    dim3 grid(batch / ROWS);                   // 4096 workgroups

    if (ws_size >= (size_t)WP_ELEMS * sizeof(float)) {
        float* Wp = (float*)d_ws;
        pad_weights<<<dim3(WP_ELEMS / 256), dim3(256), 0, stream>>>(W, Wp);
        block_linear_wmma_f32<true><<<grid, dim3(256), 0, stream>>>(x, Wp, b, out);
    } else {
        block_linear_wmma_f32<false><<<grid, dim3(256), 0, stream>>>(x, W, b, out);
    }
}